// TopKRouter_11184094839566
// MI455X (gfx1250) — compile-verified
//
#include <hip/hip_runtime.h>
#include <hip/hip_bf16.h>

// ---------------------------------------------------------------------------
// TopK expert router for MI455X (gfx1250, wave32, WMMA).
//   logits[b,e,hw] = sum_c x[b,c,hw] * W[e,c] + bias[e]
//   scores = softmax_e(logits); top-2 over e, renormalized.
// GEMM done with v_wmma_f32_16x16x4_f32 (full fp32, matches f32 reference).
// Memory-bound: x (201 MB) streamed exactly once; W (48 KB) stays in L2.
// ---------------------------------------------------------------------------

typedef float v2f __attribute__((ext_vector_type(2)));
typedef float v8f __attribute__((ext_vector_type(8)));

#define C_DIM   768      // channels (K)
#define E_DIM   16       // experts (N)
#define HW      4096     // H*W per image
#define B_DIM   16
#define PPW     32       // pixels per wave (two 16x16 WMMA tiles)
#define NWAVE   8        // waves per block
#define PPB     (PPW * NWAVE)   // 256 pixels per block

// d_out element offsets (float-sized slots)
#define OFF_PROBS   0
#define OFF_IDX     (B_DIM * 2 * HW)           // 131072
#define OFF_SCORES  (2 * B_DIM * 2 * HW)       // 262144

__global__ __launch_bounds__(256) void topk_router_kernel(
    const float* __restrict__ x,      // [B, C, H, W]
    const float* __restrict__ Wm,     // [E, C]
    const float* __restrict__ bias,   // [E]
    float* __restrict__ out)          // probs | indices(int32 bits) | scores
{
    // padded LDS tile: per-wave 32 pixels x 16 experts, 17-float rows
    // (17 coprime to 64 banks -> conflict-free transpose)
    __shared__ float lds[NWAVE][PPW][E_DIM + 1];

    const int lane = threadIdx.x & 31;
    const int wave = threadIdx.x >> 5;
    const int half = lane >> 4;   // 0: K=0,1 slice; 1: K=2,3 slice
    const int lm   = lane & 15;   // A: M (pixel-in-tile); B: N (expert)

    const int blocksPerImg = HW / PPB;                 // 16
    const int b   = blockIdx.x / blocksPerImg;
    const int hw0 = (blockIdx.x % blocksPerImg) * PPB + wave * PPW;

    // A source: x[b, 2*half + k, hw0 + lm]  (channel-major stride HW)
    const float* xa = x + ((size_t)b * C_DIM + 2 * half) * HW + hw0 + lm;
    // B source: W[lm, 2*half + k]  (row-contiguous in K -> b64 loads)
    const float* wb = Wm + (size_t)lm * C_DIM + 2 * half;

    v8f c0 = {};   // pixels hw0 +  0..15
    v8f c1 = {};   // pixels hw0 + 16..31

    #pragma unroll 4
    for (int k = 0; k < C_DIM; k += 4) {
        // B fragment: W[n, kB], W[n, kB+1] in one 64-bit load (shared by both tiles)
        v2f bv;
        {
            const float2 t = *(const float2*)wb;
            bv.x = t.x; bv.y = t.y;
        }
        // A fragments: channel kB (v0) and kB+1 (v1), pixels across lanes
        v2f a0, a1;
        a0.x = xa[0];       a0.y = xa[HW];
        a1.x = xa[16];      a1.y = xa[16 + HW];

        c0 = __builtin_amdgcn_wmma_f32_16x16x4_f32(
                 false, a0, false, bv, (short)0, c0, false, false);
        c1 = __builtin_amdgcn_wmma_f32_16x16x4_f32(
                 false, a1, false, bv, (short)0, c1, false, false);

        xa += 4 * HW;
        wb += 4;
    }

    // Lane owns expert column N = lm; D row M = v + 8*half.
    const float be = bias[lm];
    #pragma unroll
    for (int v = 0; v < 8; ++v) {
        lds[wave][v + 8 * half][lm]      = c0[v] + be;
        lds[wave][16 + v + 8 * half][lm] = c1[v] + be;
    }
    __syncthreads();

    // ---- per-pixel epilogue: softmax over 16 experts, top-2, renorm ----
    const int p  = lane;          // pixel within wave tile
    const int hw = hw0 + p;

    float s[E_DIM];
    float mx = -3.402823466e38f;
    #pragma unroll
    for (int e = 0; e < E_DIM; ++e) {
        s[e] = lds[wave][p][e];
        mx = fmaxf(mx, s[e]);
    }
    float sum = 0.0f;
    #pragma unroll
    for (int e = 0; e < E_DIM; ++e) {
        s[e] = __expf(s[e] - mx);
        sum += s[e];
    }
    const float inv = 1.0f / sum;

    float* scores_out = out + OFF_SCORES + (size_t)b * E_DIM * HW + hw;

    float v0 = -1.0f, v1 = -1.0f;
    int   i0 = 0,     i1 = 0;
    #pragma unroll
    for (int e = 0; e < E_DIM; ++e) {
        const float sc = s[e] * inv;
        scores_out[(size_t)e * HW] = sc;           // wave-contiguous per expert
        if (sc > v0)      { v1 = v0; i1 = i0; v0 = sc; i0 = e; }
        else if (sc > v1) { v1 = sc; i1 = e; }
    }

    const float rs = 1.0f / (v0 + v1);
    float* p_out = out + OFF_PROBS + (size_t)b * 2 * HW + hw;
    p_out[0]  = v0 * rs;
    p_out[HW] = v1 * rs;

    int* i_out = (int*)out + OFF_IDX + (size_t)b * 2 * HW + hw;
    i_out[0]  = i0;
    i_out[HW] = i1;
}

extern "C" void kernel_launch(void* const* d_in, const int* in_sizes, int n_in,
                              void* d_out, int out_size, void* d_ws, size_t ws_size,
                              hipStream_t stream) {
    const float* x    = (const float*)d_in[0];   // [16,768,64,64] f32
    const float* Wm   = (const float*)d_in[1];   // [16,768] f32
    const float* bias = (const float*)d_in[2];   // [16] f32
    // d_in[3] = active_experts (==2, hardcoded top-2 in kernel)
    float* out = (float*)d_out;

    const int nblocks = (B_DIM * HW) / PPB;      // 256
    topk_router_kernel<<<dim3(nblocks), dim3(256), 0, stream>>>(x, Wm, bias, out);
}